// DSSIM_43430709297529
// MI455X (gfx1250) — compile-verified
//
#include <hip/hip_runtime.h>

// DSSIM = mean((1 - SSIM)/2) over 32x3x512x512 fp32 images, 11x11 Gaussian blur.
// Fully fused single pass: x,y read once from HBM (~201MB -> ~9us floor at
// 23.3 TB/s; halo re-reads are L2 hits, L2=192MB ~ full working set).
// Separable blur as banded matmuls on the matrix cores:
//   Out(16x16) = In(16x28) x W(28x16)  via 7 chained V_WMMA_F32_16X16X4_F32
// in full fp32 (matches reference precision). One wave = one 16x16 output
// tile; 105 WMMAs per wave (5 fields x (2 row-chunks x 7 + 7)).
//
// VGPR fragment layouts per CDNA5 ISA 7.12.2:
//   A 16x4 f32 : lanes 0-15 hold (K0,K1) in (v0,v1); lanes 16-31 hold (K2,K3).
//   C/D 16x16  : VGPR v <-> row v (lanes 0-15) / row v+8 (lanes 16-31).
//   B 4x16     : assumed symmetric to A (lane n<16: rows K0,K1; n>=16: K2,K3).
//
// v2: LDS tiles physically zero-padded (row 26 = zero row, cols 26-27 = zero
// pad) so fragment builds are UNCONDITIONAL ds loads (min() clamp instead of
// branches) -> no exec-mask churn around the WMMA stream.

typedef float v2f __attribute__((ext_vector_type(2)));
typedef float v8f __attribute__((ext_vector_type(8)));

#define KS    11
#define HALO  5
#define TILE  16
#define TIN   26          // TILE + KS - 1 (valid tile extent)
#define TROWS 27          // TIN + 1 zero row
#define NK    7           // 28/4 K-steps of the 16x16x4 WMMA
#define XSTR  28          // LDS row stride (cols 26,27 are zero pad)
#define WAVES 8
#define IMG   512
#define NPLANES (32*3)
#define TILES_PER_DIM (IMG/TILE)                            // 32
#define TOTAL_TILES (NPLANES*TILES_PER_DIM*TILES_PER_DIM)   // 98304
#define NBLOCKS (TOTAL_TILES/WAVES)                         // 12288
#define SSIM_C1 (0.01f*0.01f)
#define SSIM_C2 (0.03f*0.03f)

__device__ __forceinline__ float field_val(int f, const float* xt, const float* yt,
                                           int off) {
  float xv = xt[off];
  float yv = yt[off];
  switch (f) {
    case 0:  return xv;
    case 1:  return yv;
    case 2:  return xv*xv;
    case 3:  return yv*yv;
    default: return xv*yv;
  }
}

__launch_bounds__(256, 1)
__global__ void dssim_tile_kernel(const float* __restrict__ x,
                                  const float* __restrict__ y,
                                  const float* __restrict__ kern,
                                  float* __restrict__ partial) {
  __shared__ float gk[16];                   // 1D Gaussian taps
  __shared__ float XT[WAVES][TROWS*XSTR];    // per-wave x halo tile (padded)
  __shared__ float YT[WAVES][TROWS*XSTR];    // per-wave y halo tile (padded)
  __shared__ float HB[WAVES][32*16];         // per-wave horizontal-pass buffer
  __shared__ float wsum[WAVES];

  const int tid  = threadIdx.x;
  const int wave = tid >> 5;
  const int lane = tid & 31;
  const int nloc = lane & 15;
  const bool hi  = lane >= 16;

  // Recover 1D taps: k2d = outer(g,g), sum(g)=1 => column sums of channel 0 = g
  if (tid < KS) {
    float s = 0.0f;
    #pragma unroll
    for (int i = 0; i < KS; ++i) s += kern[i*KS + tid];
    gk[tid] = s;
  }
  __syncthreads();

  // ---- tile coordinates (one wave = one 16x16 output tile) ----
  const int t     = blockIdx.x * WAVES + wave;
  const int tx    = t & (TILES_PER_DIM-1);
  const int ty    = (t >> 5) & (TILES_PER_DIM-1);
  const int plane = t >> 10;
  const int X0    = tx * TILE;
  const int Y0    = ty * TILE;
  const long base = (long)plane * IMG * IMG;

  // ---- stage 26x26 halo tiles into padded 27x28 LDS tiles (zeros elsewhere) ----
  float* xt = XT[wave];
  float* yt = YT[wave];
  for (int i = lane; i < TROWS*XSTR; i += 32) {
    int r = i / XSTR, c = i - r*XSTR;
    int gr = Y0 - HALO + r, gc = X0 - HALO + c;
    float xv = 0.0f, yv = 0.0f;
    if (r < TIN && c < TIN && gr >= 0 && gr < IMG && gc >= 0 && gc < IMG) {
      long gi = base + (long)gr*IMG + gc;
      xv = x[gi];
      yv = y[gi];
    }
    xt[i] = xv;
    yt[i] = yv;
  }
  __syncthreads();

  // ---- weight fragments (field-invariant, shared by both passes) ----
  // Horizontal B: Wc[c][n] = g[c-n]; Vertical A: WrT[m][r] = g[r-m]:
  // identical per-lane formula under the A/B layouts above.
  v2f Wf[NK];
  #pragma unroll
  for (int k = 0; k < NK; ++k) {
    int k0 = 4*k + (hi ? 2 : 0);
    int d0 = k0 - nloc, d1 = k0 + 1 - nloc;
    Wf[k].x = (d0 >= 0 && d0 < KS) ? gk[d0] : 0.0f;
    Wf[k].y = (d1 >= 0 && d1 < KS) ? gk[d1] : 0.0f;
  }

  // ---- 5 blurred fields: mu_x, mu_y, E[xx], E[yy], E[xy] ----
  v8f acc[5];
  float* hb = HB[wave];
  #pragma unroll
  for (int f = 0; f < 5; ++f) {
    // Horizontal pass: H(26x16) = In(26x28) x Wc(28x16), two 16-row chunks.
    #pragma unroll
    for (int ch = 0; ch < 2; ++ch) {
      const int m  = ch*16 + nloc;
      const int mc = (m < TIN ? m : TIN) * XSTR;   // rows >=26 -> zero row
      v8f h = {};
      #pragma unroll
      for (int k = 0; k < NK; ++k) {
        int c0 = 4*k + (hi ? 2 : 0);
        v2f a;
        a.x = field_val(f, xt, yt, mc + c0);
        a.y = field_val(f, xt, yt, mc + c0 + 1);
        h = __builtin_amdgcn_wmma_f32_16x16x4_f32(
                false, a, false, Wf[k], (short)0, h, false, false);
      }
      // scatter D -> LDS H buffer (rows r0+v / r0+v+8 per C/D layout)
      #pragma unroll
      for (int v = 0; v < 8; ++v) {
        int row = ch*16 + v + (hi ? 8 : 0);
        hb[row*16 + nloc] = h[v];
      }
    }
    __syncthreads();   // uniform across waves; H rows readable as B fragments

    // Vertical pass: V(16x16) = WrT(16x28) x H(28x16).
    v8f vacc = {};
    #pragma unroll
    for (int k = 0; k < NK; ++k) {
      int k0 = 4*k + (hi ? 2 : 0);
      v2f b;
      b.x = hb[k0*16 + nloc];
      b.y = hb[(k0+1)*16 + nloc];
      vacc = __builtin_amdgcn_wmma_f32_16x16x4_f32(
                false, Wf[k], false, b, (short)0, vacc, false, false);
    }
    acc[f] = vacc;
    __syncthreads();   // before next field overwrites HB
  }

  // ---- pointwise SSIM + local sum over this lane's 8 pixels ----
  float lsum = 0.0f;
  #pragma unroll
  for (int v = 0; v < 8; ++v) {
    float mux = acc[0][v], muy = acc[1][v];
    float sx  = acc[2][v] - mux*mux;
    float sy  = acc[3][v] - muy*muy;
    float sxy = acc[4][v] - mux*muy;
    float num = (2.0f*mux*muy + SSIM_C1) * (2.0f*sxy + SSIM_C2);
    float den = (mux*mux + muy*muy + SSIM_C1) * (sx + sy + SSIM_C2);
    float ssim = num / (den + 1e-8f);
    lsum += (1.0f - ssim) * 0.5f;
  }

  // ---- deterministic reduction: wave shuffle-tree -> block -> d_ws ----
  #pragma unroll
  for (int off = 16; off > 0; off >>= 1)
    lsum += __shfl_xor(lsum, off, 32);
  if (lane == 0) wsum[wave] = lsum;
  __syncthreads();
  if (tid == 0) {
    float s = 0.0f;
    #pragma unroll
    for (int w = 0; w < WAVES; ++w) s += wsum[w];
    partial[blockIdx.x] = s;
  }
}

__global__ void dssim_reduce_kernel(const float* __restrict__ partial,
                                    float* __restrict__ out,
                                    int n, float scale) {
  __shared__ float buf[256];
  float s = 0.0f;
  for (int i = threadIdx.x; i < n; i += 256) s += partial[i];
  buf[threadIdx.x] = s;
  __syncthreads();
  for (int o = 128; o > 0; o >>= 1) {
    if (threadIdx.x < o) buf[threadIdx.x] += buf[threadIdx.x + o];
    __syncthreads();
  }
  if (threadIdx.x == 0) out[0] = buf[0] * scale;
}

extern "C" void kernel_launch(void* const* d_in, const int* in_sizes, int n_in,
                              void* d_out, int out_size, void* d_ws, size_t ws_size,
                              hipStream_t stream) {
  const float* x    = (const float*)d_in[0];
  const float* y    = (const float*)d_in[1];
  const float* kern = (const float*)d_in[2];   // [3,1,11,11]; channel 0 used
  float* out        = (float*)d_out;
  float* partial    = (float*)d_ws;            // NBLOCKS floats (48KB) scratch

  dssim_tile_kernel<<<NBLOCKS, 256, 0, stream>>>(x, y, kern, partial);

  const float scale = 1.0f / (float)((long)NPLANES * IMG * IMG);
  dssim_reduce_kernel<<<1, 256, 0, stream>>>(partial, out, NBLOCKS, scale);
}